// myKLLoss_79731772883227
// MI455X (gfx1250) — compile-verified
//
#include <hip/hip_runtime.h>
#include <math.h>
#include <stdint.h>

typedef __attribute__((ext_vector_type(2))) float v2f;
typedef __attribute__((ext_vector_type(8))) float v8f;
typedef __attribute__((ext_vector_type(4))) unsigned int v4u;
typedef __attribute__((ext_vector_type(4))) int v4i;
typedef __attribute__((ext_vector_type(8))) int v8i;

#define NFEAT   64
#define NCLASS  10
#define AUG     80          // 64 feats + ones column + zero pad -> 5 WMMA tiles
#define CHUNK   64          // samples per staged tile
#define RSTRIDE 80          // LDS row stride in floats (TDM pad: 64 data + 16 pad)
#define GRAM_FLOATS (NCLASS * AUG * AUG)   // 64000

// ---------------------------------------------------------------------------
// Kernel 0: zero the G' accumulator region of the workspace (atomics target).
// ---------------------------------------------------------------------------
__global__ __launch_bounds__(256) void zero_ws(float* __restrict__ p, int n) {
    int i = blockIdx.x * blockDim.x + threadIdx.x;
    if (i < n) p[i] = 0.f;
}

// ---------------------------------------------------------------------------
// TDM: DMA one 64x64 f32 tile (row-major, contiguous) from global memory into
// LDS with 16 pad dwords inserted after every 64-dword row -> LDS row stride
// of 80 floats (bank-conflict-free WMMA fragment loads; pad region holds the
// constant augmented features). D# packing per CDNA5 ISA ch.8.
// ---------------------------------------------------------------------------
__device__ inline void tdm_load_tile(const float* gsrc, unsigned ldsByteAddr) {
    unsigned long long ga = (unsigned long long)(uintptr_t)gsrc;
    v4u g0;
    g0.x = 1u;                                         // count=1, user descriptor
    g0.y = ldsByteAddr;                                // lds_addr (bytes)
    g0.z = (unsigned)(ga & 0xFFFFFFFFu);               // global_addr[31:0]
    g0.w = (unsigned)((ga >> 32) & 0x01FFFFFFu)        // global_addr[56:32]
         | (2u << 30);                                 // type = 2 ("image")
    v8i g1;
    g1[0] = (int)((2u << 16)        // data_size = 4 bytes
                | (1u << 20)        // pad_enable
                | (5u << 22)        // pad_interval: 2^(5+1) = 64 dwords
                | (15u << 25));     // pad_amount: 15+1 = 16 dwords
    g1[1] = (int)(64u << 16);       // tensor_dim0 = 64
    g1[2] = (int)(64u << 16);       // tensor_dim1 = 64
    g1[3] = (int)(64u << 16);       // tile_dim0 = 64
    g1[4] = 64;                     // tile_dim1 = 64, tile_dim2 = 0
    g1[5] = 64;                     // tensor_dim0_stride = 64 elements
    g1[6] = 0;
    g1[7] = 0;
    v4i z4 = {0, 0, 0, 0};          // groups 2/3 unused (2-D tensor)
#if __clang_major__ >= 23
    v8i z8 = {0, 0, 0, 0, 0, 0, 0, 0};
    __builtin_amdgcn_tensor_load_to_lds(g0, g1, z4, z4, z8, 0);
#else
    __builtin_amdgcn_tensor_load_to_lds(g0, g1, z4, z4, 0);
#endif
}

// ---------------------------------------------------------------------------
// Kernel 1: masked augmented Gram via V_WMMA_F32_16X16X4_F32.
//   G'[c] = X~^T diag(t==c) X~,  X~ = [X | 1 | 0-pad]  (80 features)
// Block = 5 waves (160 threads). Wave m owns output row-tile m (16 rows) x 5
// column tiles. blockIdx.x = class (fast dim: the 10 class-blocks sharing a
// chunk co-reside, so the 10x logical re-read of X is served by the 192MB L2).
// Double-buffered TDM staging: wave 0 issues chunk i+1, waits tensorcnt<=1
// for chunk i, workgroup barrier fences the other waves.
// ---------------------------------------------------------------------------
__global__ __launch_bounds__(160) void gram_wmma(const float* __restrict__ X,
                                                 const int*   __restrict__ tgt,
                                                 float*       __restrict__ G,
                                                 int nchunks) {
    __shared__ float tile[2][CHUNK * RSTRIDE];   // 2 x 20KB, row-major [s][f]
    __shared__ float smask[2][CHUNK];            // per-sample class mask (0/1)

    const int cls  = blockIdx.x;
    const int tid  = threadIdx.x;
    const int wave = tid >> 5;
    const int lane = tid & 31;
    const int l    = lane & 15;            // M (A) / N (B) index within tile
    const int half = lane >> 4;            // K sub-group select
    const int mBase = wave * 16;

    // augmented features live in the TDM pad region (f=64..79) of every row:
    // f==64 -> 1.0 (sums/count row), 65..79 -> 0. TDM never writes here.
    for (int idx = tid; idx < 2 * CHUNK * 16; idx += 160) {
        int bb = idx >> 10;              // /1024
        int r  = idx & 1023;
        int s  = r >> 4;
        int f  = 64 + (r & 15);
        tile[bb][s * RSTRIDE + f] = (f == 64) ? 1.f : 0.f;
    }

    const int cnt = (nchunks - (int)blockIdx.y + (int)gridDim.y - 1) / (int)gridDim.y;
    if (cnt <= 0) return;

    // prologue: start DMA of first chunk, stage its mask
    {
        const int base0 = blockIdx.y * CHUNK;
        if (wave == 0)
            tdm_load_tile(X + (size_t)base0 * NFEAT,
                          (unsigned)(uintptr_t)(void*)&tile[0][0]);
        if (tid < CHUNK)
            smask[0][tid] = (tgt[base0 + tid] == cls) ? 1.f : 0.f;
    }

    v8f acc[5] = {};   // 5 x (16x16) f32 accumulators = 40 VGPRs

    for (int t = 0; t < cnt; ++t) {
        const int b = t & 1;
        if (t + 1 < cnt) {
            const int nbase = (blockIdx.y + (t + 1) * gridDim.y) * CHUNK;
            if (wave == 0)
                tdm_load_tile(X + (size_t)nbase * NFEAT,
                              (unsigned)(uintptr_t)(void*)&tile[1 - b][0]);
            if (tid < CHUNK)
                smask[1 - b][tid] = (tgt[nbase + tid] == cls) ? 1.f : 0.f;
        }
        if (wave == 0) {
            if (t + 1 < cnt) __builtin_amdgcn_s_wait_tensorcnt(1);
            else             __builtin_amdgcn_s_wait_tensorcnt(0);
        }
        __syncthreads();   // chunk t resident for all waves

        const float* __restrict__ buf = &tile[b][0];
        #pragma unroll 4
        for (int kk = 0; kk < CHUNK; kk += 4) {
            const int sA = kk + 2 * half;          // ISA A/B half-wave K split
            v2f a;                                  // A frag, mask applied here
            a.x = buf[sA * RSTRIDE + mBase + l]       * smask[b][sA];
            a.y = buf[(sA + 1) * RSTRIDE + mBase + l] * smask[b][sA + 1];
            #pragma unroll
            for (int n = 0; n < 5; ++n) {
                v2f bf;
                bf.x = buf[sA * RSTRIDE + n * 16 + l];
                bf.y = buf[(sA + 1) * RSTRIDE + n * 16 + l];
                acc[n] = __builtin_amdgcn_wmma_f32_16x16x4_f32(
                             false, a, false, bf, (short)0, acc[n], false, false);
            }
        }
        __syncthreads();   // compute done before buffer b is re-filled
    }

    // flush 16x80 slab: D layout row = v + 8*half, col = lane&15
    #pragma unroll
    for (int n = 0; n < 5; ++n) {
        #pragma unroll
        for (int v = 0; v < 8; ++v) {
            int row = mBase + v + 8 * half;
            int col = n * 16 + l;
            atomicAdd(&G[cls * AUG * AUG + row * AUG + col], acc[n][v]);
        }
    }
}

// ---------------------------------------------------------------------------
// Kernel 2: one block per class pair (i<=j). Builds Spair in LDS from G',
// parallel Cholesky, fwd/bwd solve for maha, logdet from the diagonal.
// (Spair[i][i] == Sc_i, so logdet_c falls out of the diagonal pairs.)
// ---------------------------------------------------------------------------
__global__ __launch_bounds__(128) void pair_kernel(const float* __restrict__ G,
                                                   float* __restrict__ mahaOut,
                                                   float* __restrict__ ldpOut) {
    __shared__ float sM[64 * 65];
    __shared__ float sMuI[64], sMuJ[64], sd[64], sy[64];
    const int tid = threadIdx.x;

    int p = blockIdx.x, i = 0;
    while (p >= NCLASS - i) { p -= NCLASS - i; ++i; }
    const int j  = i + p;
    const int bi = i * AUG * AUG, bj = j * AUG * AUG;

    const float ci  = G[bi + 64 * AUG + 64];
    const float cj  = G[bj + 64 * AUG + 64];
    const float rci = 1.f / (ci - 1.f), rcj = 1.f / (cj - 1.f);

    if (tid < 64) {
        sMuI[tid] = G[bi + 64 * AUG + tid] / ci;
        sMuJ[tid] = G[bj + 64 * AUG + tid] / cj;
    }
    __syncthreads();
    if (tid < 64) { sd[tid] = sMuI[tid] - sMuJ[tid]; sy[tid] = sd[tid]; }
    for (int idx = tid; idx < 64 * 64; idx += 128) {
        int f = idx >> 6, g = idx & 63;
        float Si = (G[bi + f * AUG + g] - ci * sMuI[f] * sMuI[g]) * rci;
        float Sj = (G[bj + f * AUG + g] - cj * sMuJ[f] * sMuJ[g]) * rcj;
        sM[f * 65 + g] = 0.5f * (Si + Sj);
    }
    __syncthreads();

    // right-looking Cholesky (lower)
    for (int k = 0; k < 64; ++k) {
        if (tid == 0) sM[k * 65 + k] = sqrtf(sM[k * 65 + k]);
        __syncthreads();
        float piv = sM[k * 65 + k];
        for (int r = k + 1 + tid; r < 64; r += 128) sM[r * 65 + k] /= piv;
        __syncthreads();
        for (int r = k + 1; r < 64; ++r) {
            float lrk = sM[r * 65 + k];
            for (int g = k + 1 + tid; g <= r; g += 128)
                sM[r * 65 + g] -= lrk * sM[g * 65 + k];
        }
        __syncthreads();
    }
    // forward solve L y = dmu
    for (int k = 0; k < 64; ++k) {
        if (tid == 0) sy[k] /= sM[k * 65 + k];
        __syncthreads();
        float yk = sy[k];
        for (int r = k + 1 + tid; r < 64; r += 128) sy[r] -= sM[r * 65 + k] * yk;
        __syncthreads();
    }
    // backward solve L^T z = y
    for (int k = 63; k >= 0; --k) {
        if (tid == 0) sy[k] /= sM[k * 65 + k];
        __syncthreads();
        float zk = sy[k];
        for (int r = tid; r < k; r += 128) sy[r] -= sM[k * 65 + r] * zk;
        __syncthreads();
    }
    if (tid == 0) {
        float dot = 0.f, ld = 0.f;
        for (int f = 0; f < 64; ++f) {
            dot += sd[f] * sy[f];
            ld  += logf(sM[f * 65 + f]);
        }
        float maha = dot * 0.125f;
        float ldp  = 2.f * ld;
        mahaOut[i * NCLASS + j] = maha; mahaOut[j * NCLASS + i] = maha;
        ldpOut[i * NCLASS + j]  = ldp;  ldpOut[j * NCLASS + i]  = ldp;
    }
}

// ---------------------------------------------------------------------------
// Kernel 3: assemble D, K, K_, two 10x10 Choleskys, final KL scalar.
// ---------------------------------------------------------------------------
__device__ inline float chol10(const float A[10][10], float L[10][10]) {
    float ld = 0.f;
    for (int k = 0; k < 10; ++k) {
        float s = A[k][k];
        for (int q = 0; q < k; ++q) s -= L[k][q] * L[k][q];
        float lkk = sqrtf(s);
        L[k][k] = lkk; ld += logf(lkk);
        for (int r = k + 1; r < 10; ++r) {
            float t = A[r][k];
            for (int q = 0; q < k; ++q) t -= L[r][q] * L[k][q];
            L[r][k] = t / lkk;
        }
    }
    return 2.f * ld;   // logdet
}

__global__ void final_kernel(const float* __restrict__ mahaIn,
                             const float* __restrict__ ldpIn,
                             const float* __restrict__ Ks,
                             float* __restrict__ out) {
    if (threadIdx.x != 0 || blockIdx.x != 0) return;
    float D[10][10], Kq[10][10], Sp[10][10], L[10][10], ldc[10];
    for (int i = 0; i < 10; ++i) ldc[i] = ldpIn[i * 10 + i];
    for (int i = 0; i < 10; ++i)
        for (int j = 0; j < 10; ++j)
            D[i][j] = mahaIn[i * 10 + j]
                    + 0.5f * (ldpIn[i * 10 + j] - 0.5f * (ldc[i] + ldc[j]));
    const float ks00 = Ks[0];
    for (int i = 0; i < 10; ++i)
        for (int j = 0; j < 10; ++j) {
            float k = expf(-(D[j][i] * D[i][j]) * 100.f);   // /0.01
            Kq[i][j] = ks00 * (0.5f * k + ((i == j) ? 0.5f : 0.f));
        }
    for (int i = 0; i < 10; ++i)
        for (int j = 0; j < 10; ++j) Sp[i][j] = Ks[i * 10 + j];
    float ldSp = chol10(Sp, L);
    // tr(Sp^{-1} K_) via one solve per column
    float tr = 0.f;
    for (int c = 0; c < 10; ++c) {
        float y[10], x[10];
        for (int r = 0; r < 10; ++r) {
            float t = Kq[r][c];
            for (int q = 0; q < r; ++q) t -= L[r][q] * y[q];
            y[r] = t / L[r][r];
        }
        for (int r = 9; r >= 0; --r) {
            float t = y[r];
            for (int q = r + 1; q < 10; ++q) t -= L[q][r] * x[q];
            x[r] = t / L[r][r];
        }
        tr += x[c];
    }
    float L2[10][10];
    float ldKq = chol10(Kq, L2);
    out[0] = 0.5f * (tr - 10.f + ldSp - ldKq);
}

// ---------------------------------------------------------------------------
extern "C" void kernel_launch(void* const* d_in, const int* in_sizes, int n_in,
                              void* d_out, int out_size, void* d_ws, size_t ws_size,
                              hipStream_t stream) {
    const float* X   = (const float*)d_in[0];
    const int*   tgt = (const int*)  d_in[1];
    const float* Ks  = (const float*)d_in[2];

    float* G    = (float*)d_ws;            // 64000 floats
    float* maha = G + GRAM_FLOATS;         // 100 floats
    float* ldp  = maha + 100;              // 100 floats

    const int N       = in_sizes[1];       // number of samples (targets)
    const int nchunks = N / CHUNK;
    int gy = nchunks < 256 ? nchunks : 256;

    zero_ws<<<(GRAM_FLOATS + 255) / 256, 256, 0, stream>>>(G, GRAM_FLOATS);
    dim3 grid(NCLASS, gy);
    gram_wmma<<<grid, 160, 0, stream>>>(X, tgt, G, nchunks);
    pair_kernel<<<NCLASS * (NCLASS + 1) / 2, 128, 0, stream>>>(G, maha, ldp);
    final_kernel<<<1, 32, 0, stream>>>(maha, ldp, Ks, (float*)d_out);
}